// Net_57363583205686
// MI455X (gfx1250) — compile-verified
//
#include <hip/hip_runtime.h>
#include <math.h>

#define TWO_PI_F 6.2831853071795864769f
#define B_SZ 8
// -0.5 * log2(e): folded into the psi features so WMMA output feeds v_exp (exp2) directly
#define NEG_HALF_LOG2E -0.72134752044448170368f

typedef __attribute__((ext_vector_type(2))) float v2f;
typedef __attribute__((ext_vector_type(8))) float v8f;

// native v_exp_f32 (computes 2^x)
__device__ __forceinline__ float fast_exp2(float x) {
  return __builtin_amdgcn_exp2f(x);
}

// ---------------------------------------------------------------------------
// K0: per-sample norm of input mixture (cov_scale_norm + weight_batch_norm,
// batch_norm=False -> stats per (b)).  Writes normalized comps; stats0={1,1}.
// ---------------------------------------------------------------------------
__global__ __launch_bounds__(128) void k_norm0(
    const float* __restrict__ in_w, const float* __restrict__ in_p,
    const float* __restrict__ in_c,
    float* __restrict__ c0w, float* __restrict__ c0p, float* __restrict__ c0c,
    float* __restrict__ stats0)
{
  const int b = blockIdx.x;
  const int t = threadIdx.x;            // 128 threads, one comp each
  __shared__ float red[128];
  __shared__ float sf, sscale;

  const int idx = b * 128 + t;
  const float w   = in_w[idx];
  const float px  = in_p[2 * idx + 0];
  const float py  = in_p[2 * idx + 1];
  const float c00 = in_c[4 * idx + 0];
  const float c01 = in_c[4 * idx + 1];
  const float c11 = in_c[4 * idx + 3];

  // mean of trace/D over the 128 components
  red[t] = 0.5f * (c00 + c11);
  __syncthreads();
  for (int s = 64; s > 0; s >>= 1) { if (t < s) red[t] += red[t + s]; __syncthreads(); }
  if (t == 0) sf = rsqrtf(red[0] / 128.0f + 1e-8f);
  __syncthreads();
  const float f = sf, f2 = f * f, f4 = f2 * f2;

  // mean |integral| with scaled covariance (det' = f^4 det)
  const float det = c00 * c11 - c01 * c01;
  red[t] = fabsf(w) * TWO_PI_F * sqrtf(fmaxf(f4 * det, 1e-12f));
  __syncthreads();
  for (int s = 64; s > 0; s >>= 1) { if (t < s) red[t] += red[t + s]; __syncthreads(); }
  if (t == 0) sscale = red[0] / 128.0f + 1e-6f;
  __syncthreads();
  const float scale = sscale;

  c0w[idx] = w / scale;
  c0p[2 * idx + 0] = px * f;
  c0p[2 * idx + 1] = py * f;
  c0c[3 * idx + 0] = c00 * f2;
  c0c[3 * idx + 1] = c01 * f2;
  c0c[3 * idx + 2] = c11 * f2;
  if (b == 0 && t == 0) { stats0[0] = 1.0f; stats0[1] = 1.0f; }
}

// ---------------------------------------------------------------------------
// Fused layer kernel: one block per (b, co).
//   A) convolve prev-GM (on-read norm f/scale) with kernel-GM; precompute
//      psi features (scaled by -0.5*log2e) -> LDS
//   B) pairwise eval via f32 WMMA 16x16x4, two row strips per wave so each
//      B fragment feeds 4 WMMAs; accumulate w_j * exp2(wmma) in registers
//   C) relu_fit weight rescale + selection score
//   D) NFIT>0: iterative top-k -> compacted output comps
//      NFIT<0 (layer 2): emit per-comp (w_new, det) + block trace sum
// ---------------------------------------------------------------------------
template <int CI, int ND, int NK, int CO, int NFIT, int N>
__global__ __launch_bounds__(256) void k_layer(
    const float* __restrict__ pw, const float* __restrict__ pp,
    const float* __restrict__ pc, const float* __restrict__ prevStats,
    const float* __restrict__ kw, const float* __restrict__ kp,
    const float* __restrict__ kc,
    float* __restrict__ ow, float* __restrict__ op, float* __restrict__ oc,
    float* __restrict__ l2w, float* __restrict__ l2d, float* __restrict__ l2T)
{
  const int t   = threadIdx.x;
  const int bid = blockIdx.x;
  const int b   = bid / CO;
  const int co  = bid % CO;

  __shared__ float sw[N], sx[N], sy[N], sca[N], scb[N], scc[N], sm[N];
  __shared__ float sPa[N], sPb2[N], sPc[N], sT3[N];   // psi (pre-scaled)
  __shared__ float rv[256];
  __shared__ int   ri[256];
  __shared__ int   sel[(NFIT > 0) ? NFIT : 1];

  // ---- Phase A: convolution of mixtures + psi precompute ---------------
  for (int n = t; n < N; n += 256) {
    const int nk = n % NK;
    const int rm = n / NK;
    const int nd = rm % ND;
    const int ci = rm / ND;

    const float fpr = prevStats[2 * ci + 0];
    const float spr = prevStats[2 * ci + 1];
    const float fp2 = fpr * fpr;

    const int pidx = (b * CI + ci) * ND + nd;
    const float wd = pw[pidx] / spr;
    const float px = pp[2 * pidx + 0] * fpr;
    const float py = pp[2 * pidx + 1] * fpr;
    const float da = pc[3 * pidx + 0] * fp2;
    const float db = pc[3 * pidx + 1] * fp2;
    const float dc = pc[3 * pidx + 2] * fp2;

    const int kidx = (co * CI + ci) * NK + nk;
    const float wk = kw[kidx];
    const float kx = kp[2 * kidx + 0];
    const float ky = kp[2 * kidx + 1];
    const float ka = kc[4 * kidx + 0];
    const float kb = kc[4 * kidx + 1];
    const float kd = kc[4 * kidx + 3];

    const float ca = da + ka, cb = db + kb, cc = dc + kd;
    const float det_d = da * dc - db * db;
    const float det_k = ka * kd - kb * kb;
    const float det_s = ca * cc - cb * cb;
    const float amp =
        TWO_PI_F * sqrtf(fmaxf(det_d * det_k / fmaxf(det_s, 1e-12f), 1e-20f));

    const float x = px + kx, y = py + ky;
    // P' = (-0.5*log2e) * inv(C): q'_ij = phi_i . psi_j = -0.5*log2e * q_ij
    const float invS = NEG_HALF_LOG2E / det_s;
    const float Pa = cc * invS, Pb = -cb * invS, Pc = ca * invS;
    const float vx = Pa * x + Pb * y;
    const float vy = Pb * x + Pc * y;

    sw[n]   = wd * wk * amp;
    sx[n]   = x;
    sy[n]   = y;
    sca[n]  = ca;
    scb[n]  = cb;
    scc[n]  = cc;
    sPa[n]  = Pa;
    sPb2[n] = 2.0f * Pb;
    sPc[n]  = Pc;
    sT3[n]  = x * vx + y * vy;
  }
  __syncthreads();

  // ---- Phase B: m(mu_i) = sum_j w_j exp(-0.5 q_ij) via WMMA ------------
  const int lane = t & 31;
  const int wv   = t >> 5;
  const int col  = lane & 15;
  const int half = lane >> 4;
  constexpr int NT = N / 16;
  constexpr int NP = NT / 2;          // process two row strips per wave

  for (int ip = wv; ip < NP; ip += 8) {
    const int i0 = ip * 32 + col;
    const int i1 = i0 + 16;
    const float x0 = sx[i0], y0 = sy[i0];
    const float x1 = sx[i1], y1 = sy[i1];
    v2f a01, a02, a11, a12;
    a01.x = half ? y0 * y0 : x0 * x0;   // A K=0 | K=2
    a01.y = half ? x0      : x0 * y0;   // A K=1 | K=3
    a02.x = half ? 0.0f    : y0;        // A K=4 | K=6(pad)
    a02.y = half ? 0.0f    : 1.0f;      // A K=5 | K=7(pad)
    a11.x = half ? y1 * y1 : x1 * x1;
    a11.y = half ? x1      : x1 * y1;
    a12.x = half ? 0.0f    : y1;
    a12.y = half ? 0.0f    : 1.0f;

    float acc0[8], acc1[8];
#pragma unroll
    for (int r = 0; r < 8; ++r) { acc0[r] = 0.0f; acc1[r] = 0.0f; }

    for (int js = 0; js < NT; ++js) {
      const int j = js * 16 + col;
      const float Pa  = sPa[j];
      const float Pb2 = sPb2[j];
      const float Pc  = sPc[j];
      const float t3  = sT3[j];
      const float xj  = sx[j], yj = sy[j];
      const float Pb  = 0.5f * Pb2;
      const float vx  = Pa * xj + Pb * yj;
      const float vy  = Pb * xj + Pc * yj;

      v2f b1, b2;
      b1.x = half ? Pc         : Pa;     // B K=0 | K=2
      b1.y = half ? -2.0f * vx : Pb2;    // B K=1 | K=3
      b2.x = half ? 0.0f       : -2.0f * vy;
      b2.y = half ? 0.0f       : t3;

      v8f c0 = {};
      c0 = __builtin_amdgcn_wmma_f32_16x16x4_f32(false, a01, false, b1,
                                                 (short)0, c0, false, false);
      c0 = __builtin_amdgcn_wmma_f32_16x16x4_f32(false, a02, false, b2,
                                                 (short)0, c0, false, false);
      v8f c1 = {};
      c1 = __builtin_amdgcn_wmma_f32_16x16x4_f32(false, a11, false, b1,
                                                 (short)0, c1, false, false);
      c1 = __builtin_amdgcn_wmma_f32_16x16x4_f32(false, a12, false, b2,
                                                 (short)0, c1, false, false);

      const float wj = sw[j];
#pragma unroll
      for (int r = 0; r < 8; ++r) {
        acc0[r] += wj * fast_exp2(c0[r]);
        acc1[r] += wj * fast_exp2(c1[r]);
      }
    }

#pragma unroll
    for (int r = 0; r < 8; ++r) {
      float v = acc0[r];
      v += __shfl_xor(v, 1);
      v += __shfl_xor(v, 2);
      v += __shfl_xor(v, 4);
      v += __shfl_xor(v, 8);
      acc0[r] = v;
      float u = acc1[r];
      u += __shfl_xor(u, 1);
      u += __shfl_xor(u, 2);
      u += __shfl_xor(u, 4);
      u += __shfl_xor(u, 8);
      acc1[r] = u;
    }
    if (col == 0) {
#pragma unroll
      for (int r = 0; r < 8; ++r) {
        sm[ip * 32 + half * 8 + r]      = acc0[r];
        sm[ip * 32 + 16 + half * 8 + r] = acc1[r];
      }
    }
  }
  __syncthreads();

  // ---- Phase C: relu_fit rescale + score -------------------------------
  for (int n = t; n < N; n += 256) {
    const float s = sm[n];
    const float denom = (fabsf(s) > 1e-6f) ? s : 1e-6f;
    const float wn = sw[n] * (fmaxf(s, 0.0f) / denom);
    sw[n] = wn;
    const float det = sca[n] * scc[n] - scb[n] * scb[n];
    sm[n] = fabsf(wn) * sqrtf(fmaxf(det, 1e-12f));  // selection score
  }
  __syncthreads();

  // ---- Phase D ---------------------------------------------------------
  if constexpr (NFIT > 0) {
    for (int it = 0; it < NFIT; ++it) {
      float bv = -1.0f; int bi = 0;
      for (int n = t; n < N; n += 256) {
        const float v = sm[n];
        if (v > bv) { bv = v; bi = n; }
      }
      rv[t] = bv; ri[t] = bi;
      __syncthreads();
      for (int s = 128; s > 0; s >>= 1) {
        if (t < s) { if (rv[t + s] > rv[t]) { rv[t] = rv[t + s]; ri[t] = ri[t + s]; } }
        __syncthreads();
      }
      if (t == 0) { sel[it] = ri[0]; sm[ri[0]] = -2.0f; }
      __syncthreads();
    }
    if (t < NFIT) {
      const int idx = sel[t];
      const int o = (b * CO + co) * NFIT + t;
      ow[o] = sw[idx];
      op[2 * o + 0] = sx[idx];
      op[2 * o + 1] = sy[idx];
      oc[3 * o + 0] = sca[idx];
      oc[3 * o + 1] = scb[idx];
      oc[3 * o + 2] = scc[idx];
    }
  } else {
    float tsum = 0.0f;
    for (int n = t; n < N; n += 256) {
      const int o = (b * CO + co) * N + n;
      l2w[o] = sw[n];
      l2d[o] = sca[n] * scc[n] - scb[n] * scb[n];
      tsum += 0.5f * (sca[n] + scc[n]);
    }
    rv[t] = tsum;
    __syncthreads();
    for (int s = 128; s > 0; s >>= 1) { if (t < s) rv[t] += rv[t + s]; __syncthreads(); }
    if (t == 0) l2T[b * CO + co] = rv[0];
  }
}

// ---------------------------------------------------------------------------
// Per-output-channel norm stats over (b, n):  stats[co] = {f, scale}.
// ---------------------------------------------------------------------------
template <int CO, int NF>
__global__ __launch_bounds__(256) void k_stats(
    const float* __restrict__ cw, const float* __restrict__ ccov,
    float* __restrict__ stats)
{
  const int co = blockIdx.x;
  const int t  = threadIdx.x;
  constexpr int M = B_SZ * NF;
  __shared__ float red[256];
  __shared__ float sf;

  float acc = 0.0f;
  for (int e = t; e < M; e += 256) {
    const int bb = e / NF, nf = e % NF;
    const int idx = (bb * CO + co) * NF + nf;
    acc += 0.5f * (ccov[3 * idx + 0] + ccov[3 * idx + 2]);
  }
  red[t] = acc;
  __syncthreads();
  for (int s = 128; s > 0; s >>= 1) { if (t < s) red[t] += red[t + s]; __syncthreads(); }
  if (t == 0) sf = rsqrtf(red[0] / (float)M + 1e-8f);
  __syncthreads();
  const float f = sf, f4 = f * f * f * f;

  acc = 0.0f;
  for (int e = t; e < M; e += 256) {
    const int bb = e / NF, nf = e % NF;
    const int idx = (bb * CO + co) * NF + nf;
    const float a = ccov[3 * idx + 0], bsy = ccov[3 * idx + 1], c = ccov[3 * idx + 2];
    const float det = a * c - bsy * bsy;
    acc += fabsf(cw[idx]) * TWO_PI_F * sqrtf(fmaxf(f4 * det, 1e-12f));
  }
  red[t] = acc;
  __syncthreads();
  for (int s = 128; s > 0; s >>= 1) { if (t < s) red[t] += red[t + s]; __syncthreads(); }
  if (t == 0) { stats[2 * co + 0] = f; stats[2 * co + 1] = red[0] / (float)M + 1e-6f; }
}

// ---------------------------------------------------------------------------
// Layer-2 final norm + integrate:  xbuf[b, co] = sum_n integrals' / scale.
// ---------------------------------------------------------------------------
__global__ __launch_bounds__(256) void k_stats2(
    const float* __restrict__ l2w, const float* __restrict__ l2d,
    const float* __restrict__ l2T, float* __restrict__ xbuf)
{
  constexpr int CO = 10, N = 1280, M = B_SZ * N;
  const int co = blockIdx.x;
  const int t  = threadIdx.x;
  __shared__ float red[256];
  __shared__ float s3[B_SZ];
  __shared__ float sf, sScale;

  if (t == 0) {
    float s = 0.0f;
    for (int bb = 0; bb < B_SZ; ++bb) s += l2T[bb * CO + co];
    sf = rsqrtf(s / (float)M + 1e-8f);
  }
  __syncthreads();
  const float f = sf, f4 = f * f * f * f;

  float aacc = 0.0f;
  for (int bb = 0; bb < B_SZ; ++bb) {
    float s = 0.0f;
    for (int k = 0; k < N / 256; ++k) {
      const int idx = (bb * CO + co) * N + k * 256 + t;
      const float term = l2w[idx] * TWO_PI_F * sqrtf(fmaxf(f4 * l2d[idx], 1e-12f));
      aacc += fabsf(term);
      s += term;
    }
    red[t] = s;
    __syncthreads();
    for (int st = 128; st > 0; st >>= 1) { if (t < st) red[t] += red[t + st]; __syncthreads(); }
    if (t == 0) s3[bb] = red[0];
    __syncthreads();
  }
  red[t] = aacc;
  __syncthreads();
  for (int st = 128; st > 0; st >>= 1) { if (t < st) red[t] += red[t + st]; __syncthreads(); }
  if (t == 0) sScale = red[0] / (float)M + 1e-6f;
  __syncthreads();
  if (t < B_SZ) xbuf[t * CO + co] = s3[t] / sScale;
}

// ---------------------------------------------------------------------------
// Final: BatchNorm1d over batch + log_softmax over classes.
// ---------------------------------------------------------------------------
__global__ __launch_bounds__(128) void k_final(
    const float* __restrict__ xbuf, float* __restrict__ out)
{
  const int t = threadIdx.x;
  __shared__ float xs[80], xn[80];
  if (t < 80) xs[t] = xbuf[t];
  __syncthreads();
  if (t < 80) {
    const int l = t % 10;
    float mu = 0.0f, m2 = 0.0f;
    for (int bb = 0; bb < 8; ++bb) {
      const float v = xs[bb * 10 + l];
      mu += v; m2 += v * v;
    }
    mu *= 0.125f; m2 *= 0.125f;
    const float var = m2 - mu * mu;
    xn[t] = (xs[t] - mu) * rsqrtf(var + 1e-5f);
  }
  __syncthreads();
  if (t < 80) {
    const int bb = t / 10;
    float mx = -3.4e38f;
    for (int l = 0; l < 10; ++l) mx = fmaxf(mx, xn[bb * 10 + l]);
    float se = 0.0f;
    for (int l = 0; l < 10; ++l) se += __expf(xn[bb * 10 + l] - mx);
    out[t] = (xn[t] - mx) - logf(se);
  }
}

// ---------------------------------------------------------------------------
extern "C" void kernel_launch(void* const* d_in, const int* in_sizes, int n_in,
                              void* d_out, int out_size, void* d_ws, size_t ws_size,
                              hipStream_t stream)
{
  (void)in_sizes; (void)n_in; (void)out_size; (void)ws_size;

  const float* in_w = (const float*)d_in[0];
  const float* in_p = (const float*)d_in[1];
  const float* in_c = (const float*)d_in[2];
  const float* k0w  = (const float*)d_in[3];
  const float* k0p  = (const float*)d_in[4];
  const float* k0c  = (const float*)d_in[5];
  const float* k1w  = (const float*)d_in[6];
  const float* k1p  = (const float*)d_in[7];
  const float* k1c  = (const float*)d_in[8];
  const float* k2w  = (const float*)d_in[9];
  const float* k2p  = (const float*)d_in[10];
  const float* k2c  = (const float*)d_in[11];

  float* ws = (float*)d_ws;
  size_t o = 0;
  float* c0w = ws + o; o += 8 * 128;
  float* c0p = ws + o; o += 8 * 128 * 2;
  float* c0c = ws + o; o += 8 * 128 * 3;
  float* st0 = ws + o; o += 2;
  float* c1w = ws + o; o += 8 * 8 * 32;
  float* c1p = ws + o; o += 8 * 8 * 32 * 2;
  float* c1c = ws + o; o += 8 * 8 * 32 * 3;
  float* st1 = ws + o; o += 16;
  float* c2w = ws + o; o += 8 * 16 * 16;
  float* c2p = ws + o; o += 8 * 16 * 16 * 2;
  float* c2c = ws + o; o += 8 * 16 * 16 * 3;
  float* st2 = ws + o; o += 32;
  float* l2w = ws + o; o += 8 * 10 * 1280;
  float* l2d = ws + o; o += 8 * 10 * 1280;
  float* l2T = ws + o; o += 80;
  float* xbf = ws + o; o += 80;

  k_norm0<<<8, 128, 0, stream>>>(in_w, in_p, in_c, c0w, c0p, c0c, st0);

  k_layer<1, 128, 5, 8, 32, 640><<<64, 256, 0, stream>>>(
      c0w, c0p, c0c, st0, k0w, k0p, k0c, c1w, c1p, c1c,
      nullptr, nullptr, nullptr);
  k_stats<8, 32><<<8, 256, 0, stream>>>(c1w, c1c, st1);

  k_layer<8, 32, 5, 16, 16, 1280><<<128, 256, 0, stream>>>(
      c1w, c1p, c1c, st1, k1w, k1p, k1c, c2w, c2p, c2c,
      nullptr, nullptr, nullptr);
  k_stats<16, 16><<<16, 256, 0, stream>>>(c2w, c2c, st2);

  k_layer<16, 16, 5, 10, -1, 1280><<<80, 256, 0, stream>>>(
      c2w, c2p, c2c, st2, k2w, k2p, k2c, nullptr, nullptr, nullptr,
      l2w, l2d, l2T);

  k_stats2<<<10, 256, 0, stream>>>(l2w, l2d, l2T, xbf);
  k_final<<<1, 128, 0, stream>>>(xbf, (float*)d_out);
}